// GILRModel_59923383714525
// MI455X (gfx1250) — compile-verified
//
#include <hip/hip_runtime.h>
#include <hip/hip_bf16.h>
#include <math.h>

// ---------------------------------------------------------------------------
// GILR model for MI455X (gfx1250, wave32, WMMA).
//   g = sigmoid(x Wg^T + bg); i = tanh(x Wi^T + bi)
//   h_t = g_t h_{t-1} + (1-g_t) i_t        (scan over T)
//   z = gelu(h W1^T + b1);  out = z W2^T + b2
// One templated GEMM kernel: C[m,n] = act(sum_k A[m,k] W[n,k] + bias[n]),
// A row-major (M x K), W row-major (N x K).
// fp32 reference -> V_WMMA_F32_16X16X4_F32 (fp32 operands + fp32 accum).
// Global->LDS staging uses CDNA5 GLOBAL_LOAD_ASYNC_TO_LDS_B128 with a
// double-buffered pipeline (ASYNCcnt-tracked, no VGPR round-trip).
// ---------------------------------------------------------------------------

typedef float v2f __attribute__((ext_vector_type(2)));
typedef float v4f __attribute__((ext_vector_type(4)));
typedef float v8f __attribute__((ext_vector_type(8)));
typedef int   v4i __attribute__((ext_vector_type(4)));

typedef __attribute__((address_space(1))) v4i* global_v4i_ptr;
typedef __attribute__((address_space(3))) v4i* lds_v4i_ptr;

#define TILE_M 128
#define TILE_N 64
#define TILE_K 32
#define LDS_STRIDE (TILE_K + 4)  // pad; keeps 16B alignment (144B row pitch)

#if __has_builtin(__builtin_amdgcn_global_load_async_to_lds_b128)
#define HAS_ASYNC_LDS 1
#else
#define HAS_ASYNC_LDS 0
#endif

__device__ __forceinline__ void async_copy16(const float* gsrc, float* ldst) {
#if HAS_ASYNC_LDS
  __builtin_amdgcn_global_load_async_to_lds_b128(
      (global_v4i_ptr)gsrc, (lds_v4i_ptr)ldst, 0, 0);
#else
  *(v4f*)ldst = *(const v4f*)gsrc;  // fallback (not expected to be taken)
#endif
}

__device__ __forceinline__ void wait_async_all() {
#if HAS_ASYNC_LDS
  asm volatile("s_wait_asynccnt 0x0" ::: "memory");
#endif
}

// issue one K-tile's worth of async copies (6 x b128 per thread)
__device__ __forceinline__ void stage_tile_async(
    const float* __restrict__ A, const float* __restrict__ W, float* AsBuf,
    float* BsBuf, int K, int mBase, int nBase, int k0, int ldRow, int ldCol) {
#pragma unroll
  for (int p = 0; p < 4; ++p) {
    const int r = ldRow + p * 32;
    async_copy16(A + (size_t)(mBase + r) * K + k0 + ldCol,
                 AsBuf + r * LDS_STRIDE + ldCol);
  }
#pragma unroll
  for (int p = 0; p < 2; ++p) {
    const int r = ldRow + p * 32;
    async_copy16(W + (size_t)(nBase + r) * K + k0 + ldCol,
                 BsBuf + r * LDS_STRIDE + ldCol);
  }
}

// 2x2 WMMA tiles per wave over one staged K-chunk
__device__ __forceinline__ void mma_chunk(const float* AsBuf, const float* BsBuf,
                                          v8f acc[2][2], int wm, int wn,
                                          int l16, int lh) {
#pragma unroll
  for (int kk = 0; kk < TILE_K; kk += 4) {
    v2f afrag[2], bfrag[2];
#pragma unroll
    for (int tm = 0; tm < 2; ++tm)
      afrag[tm] = *(const v2f*)(AsBuf + (wm * 32 + tm * 16 + l16) * LDS_STRIDE +
                                kk + 2 * lh);
#pragma unroll
    for (int tn = 0; tn < 2; ++tn)
      bfrag[tn] = *(const v2f*)(BsBuf + (wn * 32 + tn * 16 + l16) * LDS_STRIDE +
                                kk + 2 * lh);
#pragma unroll
    for (int tm = 0; tm < 2; ++tm)
#pragma unroll
      for (int tn = 0; tn < 2; ++tn)
        acc[tm][tn] = __builtin_amdgcn_wmma_f32_16x16x4_f32(
            false, afrag[tm], false, bfrag[tn], (short)0, acc[tm][tn], false,
            false);
  }
}

// ACT: 0 = none, 1 = sigmoid, 2 = tanh, 3 = exact gelu
template <int ACT>
__global__ __launch_bounds__(256)
void wmma_gemm_act(const float* __restrict__ A, const float* __restrict__ W,
                   const float* __restrict__ bias, float* __restrict__ C,
                   int M, int N, int K) {
  __shared__ __align__(16) float As[2][TILE_M * LDS_STRIDE];
  __shared__ __align__(16) float Bs[2][TILE_N * LDS_STRIDE];

  const int tid  = threadIdx.x;
  const int lane = tid & 31;
  const int wave = tid >> 5;   // 0..7
  const int wm   = wave & 3;   // 4 waves along M
  const int wn   = wave >> 2;  // 2 waves along N
  const int l16  = lane & 15;
  const int lh   = lane >> 4;

  const int mBase = blockIdx.x * TILE_M;
  const int nBase = blockIdx.y * TILE_N;

  const int ldRow = tid >> 3;       // 0..31
  const int ldCol = (tid & 7) * 4;  // 0,4,...,28

  v8f acc[2][2] = {};

#if HAS_ASYNC_LDS
  const int nk = K / TILE_K;
  stage_tile_async(A, W, As[0], Bs[0], K, mBase, nBase, 0, ldRow, ldCol);
  for (int i = 0; i < nk; ++i) {
    const int cur = i & 1;
    wait_async_all();    // this wave's copies for tile i are in LDS
    __syncthreads();     // all waves' tile-i data visible; prev readers done
    if (i + 1 < nk)      // overlap next tile's copies with compute
      stage_tile_async(A, W, As[cur ^ 1], Bs[cur ^ 1], K, mBase, nBase,
                       (i + 1) * TILE_K, ldRow, ldCol);
    mma_chunk(As[cur], Bs[cur], acc, wm, wn, l16, lh);
  }
  __syncthreads();
#else
  for (int k0 = 0; k0 < K; k0 += TILE_K) {
    // global loads first so they pipeline across the barrier
    v4f ta[4], tb[2];
#pragma unroll
    for (int p = 0; p < 4; ++p)
      ta[p] = *(const v4f*)(A + (size_t)(mBase + ldRow + p * 32) * K + k0 +
                            ldCol);
#pragma unroll
    for (int p = 0; p < 2; ++p)
      tb[p] = *(const v4f*)(W + (size_t)(nBase + ldRow + p * 32) * K + k0 +
                            ldCol);
    __syncthreads();  // previous chunk's readers done
#pragma unroll
    for (int p = 0; p < 4; ++p)
      *(v4f*)(As[0] + (ldRow + p * 32) * LDS_STRIDE + ldCol) = ta[p];
#pragma unroll
    for (int p = 0; p < 2; ++p)
      *(v4f*)(Bs[0] + (ldRow + p * 32) * LDS_STRIDE + ldCol) = tb[p];
    __syncthreads();
    mma_chunk(As[0], Bs[0], acc, wm, wn, l16, lh);
  }
#endif

  // epilogue: bias + activation + store
#pragma unroll
  for (int tm = 0; tm < 2; ++tm) {
#pragma unroll
    for (int tn = 0; tn < 2; ++tn) {
      const int n = nBase + wn * 32 + tn * 16 + l16;
      const float bv = bias[n];
#pragma unroll
      for (int r = 0; r < 8; ++r) {
        const int m = mBase + wm * 32 + tm * 16 + r + 8 * lh;
        float v = acc[tm][tn][r] + bv;
        if (ACT == 1) {
          v = 1.0f / (1.0f + __expf(-v));
        } else if (ACT == 2) {
          v = tanhf(v);
        } else if (ACT == 3) {
          v = 0.5f * v * (1.0f + erff(v * 0.70710678118654752f));
        }
        C[(size_t)m * N + n] = v;
      }
    }
  }
}

// Per-channel serial scan over time: 16384 channels (b*H + h), coalesced in h.
// h_t = g_t * h_{t-1} + (1 - g_t) * i_t
__global__ __launch_bounds__(256)
void gilr_scan(const float* __restrict__ g, const float* __restrict__ i_,
               float* __restrict__ h, int T, int BH) {
  const int c = blockIdx.x * blockDim.x + threadIdx.x;
  const float* gp = g + c;
  const float* ip = i_ + c;
  float* hp = h + c;
  float acc = 0.0f;
#pragma unroll 4
  for (int t = 0; t < T; ++t) {
    const float gv = gp[(size_t)t * BH];
    const float iv = ip[(size_t)t * BH];
    acc = gv * acc + (1.0f - gv) * iv;
    hp[(size_t)t * BH] = acc;
  }
}

extern "C" void kernel_launch(void* const* d_in, const int* in_sizes, int n_in,
                              void* d_out, int out_size, void* d_ws,
                              size_t ws_size, hipStream_t stream) {
  (void)in_sizes; (void)n_in; (void)out_size; (void)ws_size;

  const float* x  = (const float*)d_in[0];
  const float* Wg = (const float*)d_in[1];
  const float* bg = (const float*)d_in[2];
  const float* Wi = (const float*)d_in[3];
  const float* bi = (const float*)d_in[4];
  const float* W1 = (const float*)d_in[5];
  const float* b1 = (const float*)d_in[6];
  const float* W2 = (const float*)d_in[7];
  const float* b2 = (const float*)d_in[8];
  float* out = (float*)d_out;

  const int T = 2048, B = 32, IN = 256, H = 512, F = 2048, OUT = 256;
  const int M = T * B;  // 65536 rows

  // workspace layout: g | i | h | z  (128+128+128+512 MB)
  float* g  = (float*)d_ws;
  float* i_ = g + (size_t)M * H;
  float* h  = i_ + (size_t)M * H;
  float* z  = h + (size_t)M * H;

  const dim3 blk(256);

  // g = sigmoid(x Wg^T + bg)
  wmma_gemm_act<1><<<dim3(M / TILE_M, H / TILE_N), blk, 0, stream>>>(
      x, Wg, bg, g, M, H, IN);
  // i = tanh(x Wi^T + bi)
  wmma_gemm_act<2><<<dim3(M / TILE_M, H / TILE_N), blk, 0, stream>>>(
      x, Wi, bi, i_, M, H, IN);
  // h = scan(g, (1-g)*i)
  gilr_scan<<<dim3((B * H) / 256), blk, 0, stream>>>(g, i_, h, T, B * H);
  // z = gelu(h W1^T + b1)
  wmma_gemm_act<3><<<dim3(M / TILE_M, F / TILE_N), blk, 0, stream>>>(
      h, W1, b1, z, M, F, H);
  // out = z W2^T + b2
  wmma_gemm_act<0><<<dim3(M / TILE_M, OUT / TILE_N), blk, 0, stream>>>(
      z, W2, b2, out, M, OUT, F);
}